// SpectralConv2d_54700703481959
// MI455X (gfx1250) — compile-verified
//
#include <hip/hip_runtime.h>

typedef __attribute__((ext_vector_type(2))) float v2f;
typedef __attribute__((ext_vector_type(8))) float v8f;

// D(16x16,f32) = A(16x4,f32) x B(4x16,f32) + C   -- full fp32 precision path
__device__ __forceinline__ v8f wmma_f32(v2f a, v2f b, v8f c) {
  return __builtin_amdgcn_wmma_f32_16x16x4_f32(false, a, false, b, (short)0, c,
                                               false, false);
}

// ---------------------------------------------------------------------------
// Twiddle tables (computed on device each launch; deterministic, graph-safe)
//   FwRe/FwIm [256 w][32 ky]  : e^{-2pi i w ky/256}            (fwd, W axis)
//   FhRe/FhIm [64 kx][256 h]  : e^{-2pi i f(kx) h/256}         (fwd, H axis)
//   GhRe/GhIm [256 h][64 kx]  : e^{+2pi i f(kx) h/256} / 256   (inv, H axis)
//   Cw2/Sw2   [32 ky][256 w]  : Hermitian irfft fold: 1/256, 2cos/256, -2sin/256
//   f(kx) = kx for kx<32 else kx+192 (rows 224..255)
// ---------------------------------------------------------------------------
__global__ void init_twiddles_kernel(float* __restrict__ FwRe, float* __restrict__ FwIm,
                                     float* __restrict__ FhRe, float* __restrict__ FhIm,
                                     float* __restrict__ GhRe, float* __restrict__ GhIm,
                                     float* __restrict__ Cw2,  float* __restrict__ Sw2) {
  const float C = 6.283185307179586f / 256.0f;
  const int tid = blockIdx.x * blockDim.x + threadIdx.x;
  const int stride = gridDim.x * blockDim.x;
  for (int i = tid; i < 256 * 32; i += stride) {      // Fw[w][ky]
    const int w = i >> 5, ky = i & 31;
    const float th = C * (float)((w * ky) & 255);     // exact phase mod 2pi
    FwRe[i] = cosf(th);
    FwIm[i] = -sinf(th);
  }
  for (int i = tid; i < 64 * 256; i += stride) {      // Fh[kx][h]
    const int kx = i >> 8, h = i & 255;
    const int f = (kx < 32) ? kx : kx + 192;
    const float th = C * (float)((f * h) & 255);
    FhRe[i] = cosf(th);
    FhIm[i] = -sinf(th);
  }
  for (int i = tid; i < 256 * 64; i += stride) {      // Gh[h][kx], scaled 1/H
    const int h = i >> 6, kx = i & 63;
    const int f = (kx < 32) ? kx : kx + 192;
    const float th = C * (float)((f * h) & 255);
    GhRe[i] = cosf(th) * (1.0f / 256.0f);
    GhIm[i] = sinf(th) * (1.0f / 256.0f);
  }
  for (int i = tid; i < 32 * 256; i += stride) {      // Cw2/Sw2[ky][w], scaled 1/W
    const int ky = i >> 8, w = i & 255;
    if (ky == 0) {
      Cw2[i] = 1.0f / 256.0f;
      Sw2[i] = 0.0f;                                  // irfft drops Im of bin 0
    } else {
      const float th = C * (float)((ky * w) & 255);
      Cw2[i] = 2.0f * cosf(th) * (1.0f / 256.0f);
      Sw2[i] = -2.0f * sinf(th) * (1.0f / 256.0f);
    }
  }
}

// ---------------------------------------------------------------------------
// Forward truncated DFT: one workgroup per (b,ci) image.
// Stage 1: Y[256 h][32 ky] = x[h][w] * Fw[w][ky]   (A real)  -> LDS
// Stage 2: Z[64 kx][32 ky] = Fh[kx][h] * Y[h][ky]  (complex x complex)
// ---------------------------------------------------------------------------
__global__ void __launch_bounds__(256) fwd_dft_kernel(
    const float* __restrict__ x,
    const float* __restrict__ FwRe, const float* __restrict__ FwIm,
    const float* __restrict__ FhRe, const float* __restrict__ FhIm,
    float* __restrict__ Zre, float* __restrict__ Zim) {
  __shared__ float Yre[256 * 32];
  __shared__ float Yim[256 * 32];
  const int img = blockIdx.x;                       // b*64 + ci
  const float* xi = x + (size_t)img * (256 * 256);
  const int lane = threadIdx.x & 31;
  const int wave = threadIdx.x >> 5;                // 8 waves
  const int lm = lane & 15;
  const int lh = lane >> 4;

  // ---- Stage 1: 16 h-tiles, 2 per wave; both ky-tiles per pass (share A) ----
  for (int mt = 0; mt < 2; ++mt) {
    const int m0 = (wave * 2 + mt) * 16;
    v8f cr0 = {}, ci0 = {}, cr1 = {}, ci1 = {};
    for (int k0 = 0; k0 < 256; k0 += 4) {
      const int ka = k0 + 2 * lh;                   // A: reg r holds k = r + 2*(lane>>4)
      v2f a;
      a.x = xi[(m0 + lm) * 256 + ka];
      a.y = xi[(m0 + lm) * 256 + ka + 1];
      const int r0 = ka * 32, r1 = r0 + 32;         // B: reg r holds k = r + 2*(lane>>4)
      v2f br0, bi0, br1, bi1;
      br0.x = FwRe[r0 + lm];      br0.y = FwRe[r1 + lm];
      bi0.x = FwIm[r0 + lm];      bi0.y = FwIm[r1 + lm];
      br1.x = FwRe[r0 + 16 + lm]; br1.y = FwRe[r1 + 16 + lm];
      bi1.x = FwIm[r0 + 16 + lm]; bi1.y = FwIm[r1 + 16 + lm];
      cr0 = wmma_f32(a, br0, cr0);
      ci0 = wmma_f32(a, bi0, ci0);
      cr1 = wmma_f32(a, br1, cr1);
      ci1 = wmma_f32(a, bi1, ci1);
    }
#pragma unroll
    for (int r = 0; r < 8; ++r) {                   // D: reg r -> row m0 + r + 8*(lane>>4)
      const int row = m0 + r + 8 * lh;
      Yre[row * 32 + lm]      = cr0[r];
      Yim[row * 32 + lm]      = ci0[r];
      Yre[row * 32 + 16 + lm] = cr1[r];
      Yim[row * 32 + 16 + lm] = ci1[r];
    }
  }
  __syncthreads();

  // ---- Stage 2: 4 kx-tiles x 2 ky-tiles = 8 jobs, one per wave ----
  {
    const int m0 = (wave >> 1) * 16;                // kx tile
    const int n0 = (wave & 1) * 16;                 // ky tile
    v8f zrp = {}, zrm = {}, zim = {};
    for (int k0 = 0; k0 < 256; k0 += 4) {
      const int ka = k0 + 2 * lh;
      const int ab = (m0 + lm) * 256 + ka;
      v2f ar, ai;
      ar.x = FhRe[ab]; ar.y = FhRe[ab + 1];
      ai.x = FhIm[ab]; ai.y = FhIm[ab + 1];
      const int yb = ka * 32 + n0 + lm;
      v2f br, bi;
      br.x = Yre[yb]; br.y = Yre[yb + 32];
      bi.x = Yim[yb]; bi.y = Yim[yb + 32];
      zrp = wmma_f32(ar, br, zrp);                  // re: +ar*br
      zrm = wmma_f32(ai, bi, zrm);                  // re: -ai*bi (subtract at store)
      zim = wmma_f32(ar, bi, zim);                  // im: +ar*bi
      zim = wmma_f32(ai, br, zim);                  // im: +ai*br
    }
    float* zr = Zre + (size_t)img * (64 * 32);
    float* zw = Zim + (size_t)img * (64 * 32);
#pragma unroll
    for (int r = 0; r < 8; ++r) {
      const int o = (m0 + r + 8 * lh) * 32 + n0 + lm;
      zr[o] = zrp[r] - zrm[r];
      zw[o] = zim[r];
    }
  }
}

// ---------------------------------------------------------------------------
// Per-mode complex channel mixing: U[b][co] = sum_ci Z[b][ci] * Wmode[ci][co]
// M=B=16, K=CI=64, N=CO=64. Block = 2 modes, wave -> (mode half, co tile).
// Z/U layout: [b][c][kx][ky]; weights: [ci][co][32][32].
// ---------------------------------------------------------------------------
__global__ void __launch_bounds__(256) mode_mix_kernel(
    const float* __restrict__ Zre, const float* __restrict__ Zim,
    const float* __restrict__ w1re, const float* __restrict__ w1im,
    const float* __restrict__ w2re, const float* __restrict__ w2im,
    float* __restrict__ Ure, float* __restrict__ Uim) {
  const int lane = threadIdx.x & 31;
  const int wave = threadIdx.x >> 5;
  const int lm = lane & 15, lh = lane >> 4;
  const int mode = blockIdx.x * 2 + (wave >> 2);    // 0..2047
  const int n0 = (wave & 3) * 16;                   // CO tile
  const int kx = mode >> 5, ky = mode & 31;
  const float* wre;
  const float* wim;
  int wx;
  if (kx < 32) { wre = w1re; wim = w1im; wx = kx; }        // rows 0..31 -> w1
  else         { wre = w2re; wim = w2im; wx = kx - 32; }   // rows 224..255 -> w2
  const int woff = wx * 32 + ky;                    // + ci*65536 + co*1024
  const int zoff = kx * 32 + ky;                    // + b*131072 + ci*2048

  v8f urp = {}, urm = {}, uim = {};
  for (int k0 = 0; k0 < 64; k0 += 4) {
    const int ci = k0 + 2 * lh;
    const int za = lm * 131072 + ci * 2048 + zoff;  // A: m=b(lane), k=ci
    v2f ar, ai;
    ar.x = Zre[za]; ar.y = Zre[za + 2048];
    ai.x = Zim[za]; ai.y = Zim[za + 2048];
    const int wb = ci * 65536 + (n0 + lm) * 1024 + woff;  // B: k=ci, n=co(lane)
    v2f br, bi;
    br.x = wre[wb]; br.y = wre[wb + 65536];
    bi.x = wim[wb]; bi.y = wim[wb + 65536];
    urp = wmma_f32(ar, br, urp);
    urm = wmma_f32(ai, bi, urm);
    uim = wmma_f32(ar, bi, uim);
    uim = wmma_f32(ai, br, uim);
  }
#pragma unroll
  for (int r = 0; r < 8; ++r) {
    const int b = r + 8 * lh;
    const int o = b * 131072 + (n0 + lm) * 2048 + zoff;
    Ure[o] = urp[r] - urm[r];
    Uim[o] = uim[r];
  }
}

// ---------------------------------------------------------------------------
// Inverse truncated DFT: one workgroup per (b,co) image.
// Stage 1: V[256 h][32 ky] = Gh[h][kx] * U[kx][ky]  (complex)  -> LDS
// Stage 2: out[h][w] = Vre[h][ky]*Cw2[ky][w] + Vim[h][ky]*Sw2[ky][w]
// ---------------------------------------------------------------------------
__global__ void __launch_bounds__(256) inv_dft_kernel(
    const float* __restrict__ Ure, const float* __restrict__ Uim,
    const float* __restrict__ GhRe, const float* __restrict__ GhIm,
    const float* __restrict__ Cw2, const float* __restrict__ Sw2,
    float* __restrict__ out) {
  __shared__ float Vre[256 * 32];
  __shared__ float Vim[256 * 32];
  const int img = blockIdx.x;                       // b*64 + co
  const float* ur = Ure + (size_t)img * 2048;
  const float* uz = Uim + (size_t)img * 2048;
  const int lane = threadIdx.x & 31;
  const int wave = threadIdx.x >> 5;
  const int lm = lane & 15, lh = lane >> 4;

  // ---- Stage 1: 16 h-tiles, 2 per wave; both ky-tiles per pass ----
  for (int mt = 0; mt < 2; ++mt) {
    const int m0 = (wave * 2 + mt) * 16;
    v8f r0p = {}, r0m = {}, i0 = {}, r1p = {}, r1m = {}, i1 = {};
    for (int k0 = 0; k0 < 64; k0 += 4) {
      const int ka = k0 + 2 * lh;
      const int ab = (m0 + lm) * 64 + ka;
      v2f ar, ai;
      ar.x = GhRe[ab]; ar.y = GhRe[ab + 1];
      ai.x = GhIm[ab]; ai.y = GhIm[ab + 1];
      const int ub = ka * 32 + lm;
      v2f br0, bi0, br1, bi1;
      br0.x = ur[ub];      br0.y = ur[ub + 32];
      bi0.x = uz[ub];      bi0.y = uz[ub + 32];
      br1.x = ur[ub + 16]; br1.y = ur[ub + 48];
      bi1.x = uz[ub + 16]; bi1.y = uz[ub + 48];
      r0p = wmma_f32(ar, br0, r0p);
      r0m = wmma_f32(ai, bi0, r0m);
      i0  = wmma_f32(ar, bi0, i0);
      i0  = wmma_f32(ai, br0, i0);
      r1p = wmma_f32(ar, br1, r1p);
      r1m = wmma_f32(ai, bi1, r1m);
      i1  = wmma_f32(ar, bi1, i1);
      i1  = wmma_f32(ai, br1, i1);
    }
#pragma unroll
    for (int r = 0; r < 8; ++r) {
      const int row = m0 + r + 8 * lh;
      Vre[row * 32 + lm]      = r0p[r] - r0m[r];
      Vim[row * 32 + lm]      = i0[r];
      Vre[row * 32 + 16 + lm] = r1p[r] - r1m[r];
      Vim[row * 32 + 16 + lm] = i1[r];
    }
  }
  __syncthreads();

  // ---- Stage 2: 16x16 output tiles, 256 jobs round-robined over 8 waves ----
  float* op = out + (size_t)img * (256 * 256);
  for (int j = wave; j < 256; j += 8) {
    const int m0 = (j >> 4) * 16;                   // h tile
    const int n0 = (j & 15) * 16;                   // w tile
    v8f acc = {};
    for (int k0 = 0; k0 < 32; k0 += 4) {
      const int ka = k0 + 2 * lh;
      const int va = (m0 + lm) * 32 + ka;           // A from LDS
      v2f ar, ai;
      ar.x = Vre[va]; ar.y = Vre[va + 1];
      ai.x = Vim[va]; ai.y = Vim[va + 1];
      const int cb = ka * 256 + n0 + lm;
      v2f bc, bs;
      bc.x = Cw2[cb]; bc.y = Cw2[cb + 256];
      bs.x = Sw2[cb]; bs.y = Sw2[cb + 256];
      acc = wmma_f32(ar, bc, acc);
      acc = wmma_f32(ai, bs, acc);
    }
#pragma unroll
    for (int r = 0; r < 8; ++r)
      op[(m0 + r + 8 * lh) * 256 + n0 + lm] = acc[r];
  }
}

// ---------------------------------------------------------------------------
extern "C" void kernel_launch(void* const* d_in, const int* in_sizes, int n_in,
                              void* d_out, int out_size, void* d_ws, size_t ws_size,
                              hipStream_t stream) {
  const float* x    = (const float*)d_in[0];
  const float* w1re = (const float*)d_in[1];
  const float* w1im = (const float*)d_in[2];
  const float* w2re = (const float*)d_in[3];
  const float* w2im = (const float*)d_in[4];
  float* out = (float*)d_out;

  // Workspace layout (floats): 4 x 8MB planes + ~0.4MB twiddles = ~34 MB.
  float* p = (float*)d_ws;
  const size_t nZ = (size_t)16 * 64 * 64 * 32;      // [B][C][64 kx][32 ky]
  float* Zre = p;  p += nZ;
  float* Zim = p;  p += nZ;
  float* Ure = p;  p += nZ;
  float* Uim = p;  p += nZ;
  float* FwRe = p; p += 256 * 32;
  float* FwIm = p; p += 256 * 32;
  float* FhRe = p; p += 64 * 256;
  float* FhIm = p; p += 64 * 256;
  float* GhRe = p; p += 256 * 64;
  float* GhIm = p; p += 256 * 64;
  float* Cw2  = p; p += 32 * 256;
  float* Sw2  = p; p += 32 * 256;

  init_twiddles_kernel<<<128, 256, 0, stream>>>(FwRe, FwIm, FhRe, FhIm,
                                                GhRe, GhIm, Cw2, Sw2);
  fwd_dft_kernel<<<16 * 64, 256, 0, stream>>>(x, FwRe, FwIm, FhRe, FhIm, Zre, Zim);
  mode_mix_kernel<<<1024, 256, 0, stream>>>(Zre, Zim, w1re, w1im, w2re, w2im, Ure, Uim);
  inv_dft_kernel<<<16 * 64, 256, 0, stream>>>(Ure, Uim, GhRe, GhIm, Cw2, Sw2, out);

  (void)in_sizes; (void)n_in; (void)out_size; (void)ws_size;
}